// GraphAndMessages_86801289052535
// MI455X (gfx1250) — compile-verified
//
#include <hip/hip_runtime.h>
#include <hip/hip_bf16.h>
#include <math.h>

// ---------------------------------------------------------------------------
// GraphAndMessages fused edge-MLP for gfx1250 (CDNA5, wave32, WMMA f32)
// ---------------------------------------------------------------------------
// pairs(E,32) = [inputs[src], inputs[snk]]
// h = pairs@W0 + b0                          (pass 1: column sum/sumsq -> ws)
// h = (h-mean)*rsqrt(var+eps)*gamma + beta   (finalize: scale/shift -> ws)
// h = lrelu(h)@W1 + b1 ; out = sigmoid(lrelu(h)@W2 + b2)   (pass 2, fused)
// edges = out[0::2], messages = out[1::2], concatenated flat in d_out.
// ---------------------------------------------------------------------------

typedef float v2f __attribute__((ext_vector_type(2)));
typedef float v8f __attribute__((ext_vector_type(8)));

#define NUM_OBJ 1500
#define OBJ_DIM 16
#define E_EDGES (NUM_OBJ * (NUM_OBJ - 1))   // 2,248,500
#define EPS_BN  1e-5f
#define SLOPE   0.01f
#define N_TILES ((E_EDGES + 15) / 16)       // 140,532
#define WAVES_PER_BLOCK 8

// ws layout (floats): [0..31] colsum  [32..63] colsumsq
//                     [64..95] scale  [96..127] shift
__global__ __launch_bounds__(64) void gm_init(float* ws) {
    if (threadIdx.x < 64) ws[threadIdx.x] = 0.0f;
}

__device__ __forceinline__ void load_wfrag(const float* __restrict__ W,
                                           int m16, int hi, v2f wf[2][8]) {
    #pragma unroll
    for (int g = 0; g < 2; ++g) {
        #pragma unroll
        for (int ks = 0; ks < 8; ++ks) {
            const int k = ks * 4 + 2 * hi;
            const int n = g * 16 + m16;
            wf[g][ks].x = W[k * 32 + n];
            wf[g][ks].y = W[(k + 1) * 32 + n];
        }
    }
}

// Gather the A-fragments (16 rows x 32 cols of "pairs") for this wave's tile.
__device__ __forceinline__ void load_afrag(const float* __restrict__ inputs,
                                           const int* __restrict__ src_idx,
                                           const int* __restrict__ snk_idx,
                                           int r0, int m16, int hi, v2f a[8]) {
    int r = r0 + m16;
    if (r >= E_EDGES) r = E_EDGES - 1;              // clamp (masked later)
    const int so = src_idx[r] * OBJ_DIM;
    const int ko = snk_idx[r] * OBJ_DIM;
    #pragma unroll
    for (int ks = 0; ks < 8; ++ks) {
        const int c = ks * 4 + 2 * hi;              // column in [0,32)
        const float* p = (c < 16) ? (inputs + so + c) : (inputs + ko + (c - 16));
        a[ks].x = p[0];
        a[ks].y = p[1];
    }
}

// 16x32 @ 32x32 via 8 K-steps x 2 N-tiles of V_WMMA_F32_16X16X4_F32
__device__ __forceinline__ void mm_layer(const v2f a[8], const v2f wf[2][8],
                                         v8f& c0, v8f& c1) {
    #pragma unroll
    for (int ks = 0; ks < 8; ++ks) {
        c0 = __builtin_amdgcn_wmma_f32_16x16x4_f32(false, a[ks], false, wf[0][ks],
                                                   (short)0, c0, false, false);
        c1 = __builtin_amdgcn_wmma_f32_16x16x4_f32(false, a[ks], false, wf[1][ks],
                                                   (short)0, c1, false, false);
    }
}

// ---------------------------------------------------------------------------
// Pass 1: h = pairs@W0 + b0; accumulate per-column sum / sumsq into ws[0..63]
// ---------------------------------------------------------------------------
__global__ __launch_bounds__(256) void gm_pass1(const float* __restrict__ inputs,
                                                const int*   __restrict__ src_idx,
                                                const int*   __restrict__ snk_idx,
                                                const float* __restrict__ W0,
                                                const float* __restrict__ b0,
                                                float* __restrict__ ws) {
    __shared__ float red[64];
    const int tid  = threadIdx.x;
    const int lane = tid & 31;
    const int wave = tid >> 5;
    const int m16  = lane & 15;
    const int hi   = lane >> 4;

    if (tid < 64) red[tid] = 0.0f;
    __syncthreads();

    v2f w0f[2][8];
    load_wfrag(W0, m16, hi, w0f);
    const float b0v0 = b0[m16];
    const float b0v1 = b0[16 + m16];

    const int tile = blockIdx.x * WAVES_PER_BLOCK + wave;
    const int r0   = tile * 16;

    v2f a[8];
    load_afrag(inputs, src_idx, snk_idx, r0, m16, hi, a);

    v8f c0 = {}, c1 = {};
    mm_layer(a, w0f, c0, c1);

    // valid rows in this half-wave's M-range [8*hi, 8*hi+8)
    int left = E_EDGES - r0 - 8 * hi;
    if (left < 0) left = 0;
    if (left > 8) left = 8;

    float s0 = 0.f, q0 = 0.f, s1 = 0.f, q1 = 0.f;
    #pragma unroll
    for (int v = 0; v < 8; ++v) {
        if (v < left) {
            const float h0 = c0[v] + b0v0;
            const float h1 = c1[v] + b0v1;
            s0 += h0; q0 += h0 * h0;
            s1 += h1; q1 += h1 * h1;
        }
    }
    atomicAdd(&red[m16],      s0);   // col = m16
    atomicAdd(&red[32 + m16], q0);
    atomicAdd(&red[16 + m16], s1);   // col = 16 + m16
    atomicAdd(&red[48 + m16], q1);
    __syncthreads();
    if (tid < 64) atomicAdd(&ws[tid], red[tid]);
}

// ---------------------------------------------------------------------------
// Finalize batch-norm constants: scale = gamma*rsqrt(var+eps),
//                                shift = beta - mean*scale
// ---------------------------------------------------------------------------
__global__ __launch_bounds__(32) void gm_finalize(const float* __restrict__ gamma,
                                                  const float* __restrict__ beta,
                                                  float* __restrict__ ws) {
    const int c = threadIdx.x;
    if (c < 32) {
        const float inv_n = 1.0f / (float)E_EDGES;
        const float mean  = ws[c] * inv_n;
        const float var   = ws[32 + c] * inv_n - mean * mean;
        const float scale = gamma[c] * rsqrtf(var + EPS_BN);
        ws[64 + c] = scale;
        ws[96 + c] = beta[c] - mean * scale;
    }
}

// ---------------------------------------------------------------------------
// Pass 2: recompute h, normalize, lrelu, @W1+b1, lrelu, @W2+b2, sigmoid,
//         de-interleave into d_out = [edges (E/2 x 2) | messages (E/2 x 2)]
// ---------------------------------------------------------------------------
__global__ __launch_bounds__(256) void gm_pass2(const float* __restrict__ inputs,
                                                const int*   __restrict__ src_idx,
                                                const int*   __restrict__ snk_idx,
                                                const float* __restrict__ W0,
                                                const float* __restrict__ b0,
                                                const float* __restrict__ W1,
                                                const float* __restrict__ b1,
                                                const float* __restrict__ W2,
                                                const float* __restrict__ b2,
                                                const float* __restrict__ ws,
                                                float* __restrict__ out) {
    // per-wave 16x32 staging tile, stride 33 to avoid LDS bank conflicts
    __shared__ float st[WAVES_PER_BLOCK * 16 * 33];

    const int tid  = threadIdx.x;
    const int lane = tid & 31;
    const int wave = tid >> 5;
    const int m16  = lane & 15;
    const int hi   = lane >> 4;
    const int base = wave * 16 * 33;

    v2f w0f[2][8], w1f[2][8];
    load_wfrag(W0, m16, hi, w0f);
    load_wfrag(W1, m16, hi, w1f);
    const float b0v0 = b0[m16],      b0v1 = b0[16 + m16];
    const float b1v0 = b1[m16],      b1v1 = b1[16 + m16];
    const float sc0  = ws[64 + m16], sc1  = ws[64 + 16 + m16];
    const float sh0  = ws[96 + m16], sh1  = ws[96 + 16 + m16];

    const int tile = blockIdx.x * WAVES_PER_BLOCK + wave;
    const int r0   = tile * 16;

    // ---- layer 0 (recompute h) ----
    v2f a[8];
    load_afrag(inputs, src_idx, snk_idx, r0, m16, hi, a);
    v8f c0 = {}, c1 = {};
    mm_layer(a, w0f, c0, c1);

    // normalize + lrelu, stage to LDS in row-major for re-layout to A-frags
    #pragma unroll
    for (int v = 0; v < 8; ++v) {
        const int m = v + 8 * hi;
        float h0 = (c0[v] + b0v0) * sc0 + sh0;
        float h1 = (c1[v] + b0v1) * sc1 + sh1;
        h0 = (h0 >= 0.f) ? h0 : SLOPE * h0;
        h1 = (h1 >= 0.f) ? h1 : SLOPE * h1;
        st[base + m * 33 + m16]      = h0;
        st[base + m * 33 + 16 + m16] = h1;
    }
    __syncthreads();

    // ---- layer 1 ----
    #pragma unroll
    for (int ks = 0; ks < 8; ++ks) {
        const int c = ks * 4 + 2 * hi;
        a[ks].x = st[base + m16 * 33 + c];
        a[ks].y = st[base + m16 * 33 + c + 1];
    }
    v8f d0 = {}, d1 = {};
    mm_layer(a, w1f, d0, d1);

    __syncthreads();   // everyone done reading before overwriting staging tile

    #pragma unroll
    for (int v = 0; v < 8; ++v) {
        const int m = v + 8 * hi;
        float h0 = d0[v] + b1v0;
        float h1 = d1[v] + b1v1;
        h0 = (h0 >= 0.f) ? h0 : SLOPE * h0;
        h1 = (h1 >= 0.f) ? h1 : SLOPE * h1;
        st[base + m * 33 + m16]      = h0;
        st[base + m * 33 + 16 + m16] = h1;
    }
    __syncthreads();

    // ---- layer 2 head (32 -> 2) + sigmoid + scatter ----
    // lane = m16 handles row m16, output channel j = hi
    const int j = hi;
    float acc = b2[j];
    #pragma unroll
    for (int c = 0; c < 32; ++c)
        acc = fmaf(st[base + m16 * 33 + c], W2[c * 2 + j], acc);
    acc = 1.0f / (1.0f + __expf(-acc));

    const int r = r0 + m16;
    if (r < E_EDGES) {
        // even rows -> edges block [0, E); odd rows -> messages block [E, 2E)
        const int idx = ((r & 1) ? E_EDGES : 0) + (r >> 1) * 2 + j;
        out[idx] = acc;
    }
}

extern "C" void kernel_launch(void* const* d_in, const int* in_sizes, int n_in,
                              void* d_out, int out_size, void* d_ws, size_t ws_size,
                              hipStream_t stream) {
    const float* inputs  = (const float*)d_in[0];
    const int*   src_idx = (const int*)  d_in[1];
    const int*   snk_idx = (const int*)  d_in[2];
    const float* W0      = (const float*)d_in[3];
    const float* b0      = (const float*)d_in[4];
    const float* gamma   = (const float*)d_in[5];
    const float* beta    = (const float*)d_in[6];
    const float* W1      = (const float*)d_in[7];
    const float* b1      = (const float*)d_in[8];
    const float* W2      = (const float*)d_in[9];
    const float* b2      = (const float*)d_in[10];
    float* out = (float*)d_out;
    float* ws  = (float*)d_ws;

    const int blocks = (N_TILES + WAVES_PER_BLOCK - 1) / WAVES_PER_BLOCK;

    gm_init<<<1, 64, 0, stream>>>(ws);
    gm_pass1<<<blocks, 256, 0, stream>>>(inputs, src_idx, snk_idx, W0, b0, ws);
    gm_finalize<<<1, 32, 0, stream>>>(gamma, beta, ws);
    gm_pass2<<<blocks, 256, 0, stream>>>(inputs, src_idx, snk_idx,
                                         W0, b0, W1, b1, W2, b2, ws, out);
}